// Patch_Embed_Occlude_64433099374846
// MI455X (gfx1250) — compile-verified
//
#include <hip/hip_runtime.h>

// Patch occlusion: out[b,c,h,w] = x[b,c,h,w] * (mask[b, w/16, h/16] >= 0.5 ? 1 : 0)
// Pure streaming kernel: HBM-bound (~226 MB traffic, roofline ~10us @ 23.3 TB/s).
// Strategy: b128 non-temporal loads/stores, zero per-element div/mod via grid mapping.

typedef __attribute__((ext_vector_type(4))) float v4f;

namespace {
constexpr int C  = 3;
constexpr int H  = 384;
constexpr int W  = 384;
constexpr int PS = 16;
constexpr int Hg = H / PS;   // 24
constexpr int Wg = W / PS;   // 24
constexpr int W4 = W / 4;    // 96 float4 per row
}

// block = (96, 4): tx = float4 column (covers full row), ty = 4-row group in patch.
// grid  = (Hg, C, B): one block per (patch-row-stripe, channel, batch).
__global__ __launch_bounds__(384) void Patch_Embed_Occlude_kernel(
    const float* __restrict__ x,
    const float* __restrict__ mask,
    float* __restrict__ out)
{
    const int tx = threadIdx.x;      // 0..95
    const int ty = threadIdx.y;      // 0..3
    const int hg = blockIdx.x;       // 0..Hg-1
    const int c  = blockIdx.y;       // 0..C-1
    const int b  = blockIdx.z;       // 0..B-1

    // One mask value covers this thread's entire 4-row x 4-col float4 strip:
    // wg = tx/4 (16 floats per patch = 4 float4s), hg fixed per block,
    // rows h0..h0+3 stay inside patch row-group since 4 | 16.
    const int wg = tx >> 2;
    const float m = mask[(size_t)b * (Wg * Hg) + wg * Hg + hg];
    const float keep = (m >= 0.5f) ? 1.0f : 0.0f;

    const int h0 = hg * PS + ty * 4;
    const size_t base = ((size_t)(b * C + c) * H + h0) * W4 + tx;  // float4 units

    const v4f* __restrict__ xin = (const v4f*)x;
    v4f* __restrict__ xo        = (v4f*)out;

    // Issue all loads first (MLP), then multiply + NT stores.
    v4f v0 = __builtin_nontemporal_load(xin + base + 0 * (size_t)W4);
    v4f v1 = __builtin_nontemporal_load(xin + base + 1 * (size_t)W4);
    v4f v2 = __builtin_nontemporal_load(xin + base + 2 * (size_t)W4);
    v4f v3 = __builtin_nontemporal_load(xin + base + 3 * (size_t)W4);

    v0 *= keep;
    v1 *= keep;
    v2 *= keep;
    v3 *= keep;

    __builtin_nontemporal_store(v0, xo + base + 0 * (size_t)W4);
    __builtin_nontemporal_store(v1, xo + base + 1 * (size_t)W4);
    __builtin_nontemporal_store(v2, xo + base + 2 * (size_t)W4);
    __builtin_nontemporal_store(v3, xo + base + 3 * (size_t)W4);
}

extern "C" void kernel_launch(void* const* d_in, const int* in_sizes, int n_in,
                              void* d_out, int out_size, void* d_ws, size_t ws_size,
                              hipStream_t stream) {
    const float* x    = (const float*)d_in[0];
    const float* mask = (const float*)d_in[1];
    float* out        = (float*)d_out;

    // Derive batch from mask element count (B * Wg * Hg).
    const int B = in_sizes[1] / (Wg * Hg);

    dim3 block(W4, 4);       // 96 x 4 = 384 threads = 12 wave32
    dim3 grid(Hg, C, B);     // 24 x 3 x B

    Patch_Embed_Occlude_kernel<<<grid, block, 0, stream>>>(x, mask, out);
}